// LaplacianSmoothLoss_31928786878950
// MI455X (gfx1250) — compile-verified
//
#include <hip/hip_runtime.h>

// Problem constants (fixed by setup_inputs): V=8192 vertices, F=16384 faces.
#define V_N 8192
#define F_N 16384
#define WORDS_PER_ROW 256                       // 8192 bits / 32
#define BM_WORDS (V_N * WORDS_PER_ROW)          // 2M words = 8 MB bitmask
#define LVERTS_BYTES (V_N * 4 * sizeof(float))  // 128 KB padded vertex table

typedef __attribute__((ext_vector_type(2))) float v2f;
typedef __attribute__((ext_vector_type(8))) float v8f;

// ---------------------------------------------------------------------------
// Kernel 1: zero the adjacency bitmask and the scalar output accumulator.
// ---------------------------------------------------------------------------
__global__ void zero_ws_kernel(unsigned* __restrict__ bm, float* __restrict__ out) {
    int i = blockIdx.x * blockDim.x + threadIdx.x;
    int stride = gridDim.x * blockDim.x;
    for (int w = i; w < BM_WORDS; w += stride) bm[w] = 0u;
    if (i == 0) out[0] = 0.0f;
}

// ---------------------------------------------------------------------------
// Kernel 2: scatter edge bits (set semantics via atomicOr).
// ---------------------------------------------------------------------------
__global__ void build_adj_kernel(const int* __restrict__ faces,
                                 unsigned* __restrict__ bm) {
    int i = blockIdx.x * blockDim.x + threadIdx.x;
    if (i < F_N * 6) {
        const int SRC[6] = {0, 0, 1, 1, 2, 2};
        const int DST[6] = {1, 2, 0, 2, 0, 1};
        int f = i / 6, p = i % 6;
        int r = faces[f * 3 + SRC[p]];
        int c = faces[f * 3 + DST[p]];
        atomicOr(&bm[r * WORDS_PER_ROW + (c >> 5)], 1u << (c & 31));
    }
}

// ---------------------------------------------------------------------------
// Kernel 3: 256 threads = 8 waves per block; 2 tiles x 4 K-slabs.
// Each wave computes a partial D(16x16) over a 2048-column slab:
//   D += A(16x4, bits->f32) x B(4x16, rows = [x y z 1 0..0])
// via V_WMMA_F32_16X16X4_F32 (exact fp32). D[:,0..2] = A @ verts,
// D[:,3] = degree. Wave-uniform ballot skips at 32-col and 4-col granularity
// (scalar branches keep EXEC all-1s for WMMA). Bitmask rows are streamed as
// b128 with a 4-deep rolling prefetch; B fragments come from a 128 KB LDS
// vertex table. Partial D's combine through LDS across the 4 K-slab waves.
// ---------------------------------------------------------------------------
__global__ void __launch_bounds__(256)
lap_loss_kernel(const unsigned* __restrict__ bm,
                const float* __restrict__ verts,
                float* __restrict__ out) {
    extern __shared__ float lverts[];            // V_N x float4 (x,y,z,1)
    __shared__ float spill[8][64];               // per-wave 16 rows x 4 values

    const int lane      = threadIdx.x & 31;
    const int waveInBlk = threadIdx.x >> 5;      // 0..7
    const int slab      = waveInBlk & 3;         // K slab: 2048 columns
    const int tileInBlk = waveInBlk >> 2;        // 0..1
    const int tile      = blockIdx.x * 2 + tileInBlk;   // 0..511
    const int rowBase   = tile * 16;
    const int m         = lane & 15;             // row within tile (both halves)
    const bool hi       = lane >= 16;            // fragment half (K offset +2)
    const int n         = lane & 15;             // B/C/D column
    const int nc        = n & 3;                 // clamped column for LDS read
    const bool nActive  = (n < 4);

    // ---- Fill the padded vertex table in LDS (once per block). ----
    for (int v = threadIdx.x; v < V_N; v += 256) {
        lverts[v * 4 + 0] = verts[v * 3 + 0];
        lverts[v * 4 + 1] = verts[v * 3 + 1];
        lverts[v * 4 + 2] = verts[v * 3 + 2];
        lverts[v * 4 + 3] = 1.0f;                // degree column
    }
    __syncthreads();

    v8f c = {0.f, 0.f, 0.f, 0.f, 0.f, 0.f, 0.f, 0.f};

    // This wave's 16 b128 groups of its row's bitmask (slab = 2048 columns).
    const uint4* __restrict__ rp =
        (const uint4*)bm + (size_t)(rowBase + m) * (WORDS_PER_ROW / 4) + slab * 16;

    uint4 buf[4];
    #pragma unroll
    for (int p = 0; p < 4; ++p) buf[p] = rp[p];  // 4-deep prefetch prologue

    #pragma unroll 4
    for (int g = 0; g < 16; ++g) {               // 16 x 128 columns
        uint4 cur = buf[g & 3];
        int gp = g + 4;
        buf[g & 3] = rp[(gp < 16) ? gp : 15];    // clamped rolling prefetch

        #pragma unroll
        for (int q = 0; q < 4; ++q) {
            unsigned w = (q == 0) ? cur.x : (q == 1) ? cur.y
                       : (q == 2) ? cur.z : cur.w;
            // Any bit set in these 32 columns across the 16 rows? (uniform)
            if (__builtin_amdgcn_ballot_w32(w != 0u) == 0u) continue;

            const int kBase = (slab * 16 + g) * 128 + q * 32;
            #pragma unroll
            for (int j = 0; j < 8; ++j) {        // 8 chunks of K=4
                unsigned nib = (w >> (j * 4)) & 0xFu;
                if (__builtin_amdgcn_ballot_w32(nib != 0u) == 0u) continue;

                // A fragment: VGPR0 = K{0,2}, VGPR1 = K{1,3}
                int sh = hi ? 2 : 0;
                v2f a, b;
                a.x = (float)((nib >> sh) & 1u);
                a.y = (float)((nib >> (sh + 1)) & 1u);
                // B fragment from LDS: row k = (x,y,z,1,0,...,0)
                int kv = kBase + j * 4 + (hi ? 2 : 0);
                float t0 = lverts[kv * 4 + nc];
                float t1 = lverts[(kv + 1) * 4 + nc];
                b.x = nActive ? t0 : 0.0f;
                b.y = nActive ? t1 : 0.0f;
                c = __builtin_amdgcn_wmma_f32_16x16x4_f32(
                        false, a, false, b, (short)0, c, false, false);
            }
        }
    }

    // ---- Spill partial D[:,0..3]: VGPR r = M=r (lanes 0-15) / M=r+8. ----
    if (nActive) {
        #pragma unroll
        for (int r = 0; r < 8; ++r) {
            int row = r + (hi ? 8 : 0);
            spill[waveInBlk][row * 4 + n] = c[r];
        }
    }
    __syncthreads();

    // ---- Combine 4 K-slab partials and finalize (waves 0 and 4 only). ----
    float part = 0.0f;
    if ((waveInBlk & 3) == 0 && !hi) {           // lanes 0..15: one row each
        int row = rowBase + lane;
        float Dx = 0.f, Dy = 0.f, Dz = 0.f, dg = 0.f;
        #pragma unroll
        for (int p = 0; p < 4; ++p) {
            Dx += spill[tileInBlk * 4 + p][lane * 4 + 0];
            Dy += spill[tileInBlk * 4 + p][lane * 4 + 1];
            Dz += spill[tileInBlk * 4 + p][lane * 4 + 2];
            dg += spill[tileInBlk * 4 + p][lane * 4 + 3];
        }
        float vx = lverts[row * 4 + 0];
        float vy = lverts[row * 4 + 1];
        float vz = lverts[row * 4 + 2];
        float lx = dg * vx - Dx;
        float ly = dg * vy - Dy;
        float lz = dg * vz - Dz;
        part = lx * lx + ly * ly + lz * lz;
    }
    part += __shfl_xor(part, 1, 32);
    part += __shfl_xor(part, 2, 32);
    part += __shfl_xor(part, 4, 32);
    part += __shfl_xor(part, 8, 32);
    part += __shfl_xor(part, 16, 32);
    if (lane == 0 && (waveInBlk & 3) == 0)
        atomicAdd(out, part * (0.1f / (float)V_N));   // WEIGHT * mean
}

// ---------------------------------------------------------------------------
extern "C" void kernel_launch(void* const* d_in, const int* in_sizes, int n_in,
                              void* d_out, int out_size, void* d_ws, size_t ws_size,
                              hipStream_t stream) {
    const float* verts = (const float*)d_in[0];   // (1, 8192, 3) f32
    const int*   faces = (const int*)d_in[1];     // (16384, 3) i32
    float* out = (float*)d_out;                   // scalar f32

    unsigned* bm = (unsigned*)d_ws;               // 8 MB bitmask

    zero_ws_kernel<<<2048, 256, 0, stream>>>(bm, out);
    build_adj_kernel<<<(F_N * 6 + 255) / 256, 256, 0, stream>>>(faces, bm);
    lap_loss_kernel<<<(V_N / 16) / 2, 256, LVERTS_BYTES, stream>>>(bm, verts, out);
}